// Attention_layer_56590489092432
// MI455X (gfx1250) — compile-verified
//
#include <hip/hip_runtime.h>
#include <hip/hip_bf16.h>

typedef __attribute__((ext_vector_type(2))) float v2f;
typedef __attribute__((ext_vector_type(8))) float v8f;

#if defined(__AMDGCN__) && !__has_builtin(__builtin_amdgcn_wmma_f32_16x16x4_f32)
#warning "CDNA5 f32 16x16x4 WMMA builtin NOT available on this toolchain"
#endif

static __device__ __forceinline__ v8f wmma4(v2f a, v2f b, v8f c) {
#if defined(__AMDGCN__) && __has_builtin(__builtin_amdgcn_wmma_f32_16x16x4_f32)
  return __builtin_amdgcn_wmma_f32_16x16x4_f32(false, a, false, b, (short)0, c,
                                               false, false);
#else
  c[0] += a[0] * b[0];  // host-pass placeholder
  return c;
#endif
}

static __device__ __forceinline__ v2f ld2(const float* p) {
  return *(const v2f*)p;
}

// Async gather into LDS: one wave-level instruction moves 32 lanes x 16B.
// LDS offset = low 32 bits of the generic shared-aperture address (ISA 10.2).
static __device__ __forceinline__ void async_copy16(const float* g, float* l) {
  unsigned lds = (unsigned)(uintptr_t)l;
  unsigned long long ga = (unsigned long long)(uintptr_t)g;
  asm volatile("global_load_async_to_lds_b128 %0, %1, off"
               :
               : "v"(lds), "v"(ga)
               : "memory");
}
static __device__ __forceinline__ void async_wait0() {
  asm volatile("s_wait_asynccnt 0x0" ::: "memory");
}

// ---------------------------------------------------------------------------
#define BB 16
#define SS_ 512
#define HH 16
#define DH 64
#define DD 1024

// ---------------------------------------------------------------------------
// Kernel 1: per-head QKV projection.  C[8192, 64] = X[8192,1024] @ W[h].
// grid = (64 Mtiles, 16 heads, 3 {q,k,v}); 256 threads = 8 waves.
// A tile staged by async-DMA; B tile staged TRANSPOSED for contiguous frags.
// ---------------------------------------------------------------------------
__global__ __launch_bounds__(256) void proj_wmma_kernel(
    const float* __restrict__ X, const float* __restrict__ Wq,
    const float* __restrict__ Wk, const float* __restrict__ Wv,
    float* __restrict__ Qo, float* __restrict__ Ko, float* __restrict__ Vo) {
  const int mt = blockIdx.x;
  const int h = blockIdx.y;
  const int which = blockIdx.z;

  const float* W = (which == 0) ? Wq : (which == 1) ? Wk : Wv;
  W += (size_t)h * DD * DH;
  float* O = (which == 0) ? Qo : (which == 1) ? Ko : Vo;

  __shared__ __align__(16) float As[128][20];   // 128 x 16 (K), stride 20
  __shared__ __align__(16) float BsT[64][20];   // transposed: BsT[n][k]

  const int tid = threadIdx.x;
  const int wave = tid >> 5, lane = tid & 31;
  const int hi = lane >> 4, lo = lane & 15;
  const int m0 = wave * 16;

  v8f acc[4] = {};
  const float* Arow = X + (size_t)mt * 128 * DD;

  for (int k0 = 0; k0 < DD; k0 += 16) {
    __syncthreads();
    // A tile: 128x16 floats = 512 x 16B chunks, 2 per thread, async DMA
    for (int i = tid; i < 512; i += 256) {
      int r = i >> 2, c4 = (i & 3) << 2;
      async_copy16(Arow + (size_t)r * DD + k0 + c4, &As[r][c4]);
    }
    // B tile 16x64, transposed into BsT[n][k]
    {
      int r = tid >> 4, c4 = (tid & 15) << 2;
      float4 w4 = *(const float4*)(W + (size_t)(k0 + r) * DH + c4);
      BsT[c4 + 0][r] = w4.x;
      BsT[c4 + 1][r] = w4.y;
      BsT[c4 + 2][r] = w4.z;
      BsT[c4 + 3][r] = w4.w;
    }
    async_wait0();
    __syncthreads();

    for (int kk = 0; kk < 16; kk += 4) {
      v2f a = ld2(&As[m0 + lo][kk + 2 * hi]);
#pragma unroll
      for (int n = 0; n < 4; ++n) {
        v2f b = ld2(&BsT[n * 16 + lo][kk + 2 * hi]);
        acc[n] = wmma4(a, b, acc[n]);
      }
    }
  }

#pragma unroll
  for (int n = 0; n < 4; ++n)
#pragma unroll
    for (int v = 0; v < 8; ++v) {
      int gm = mt * 128 + m0 + v + 8 * hi;
      int b_ = gm >> 9, s_ = gm & (SS_ - 1);
      size_t off = (((size_t)b_ * HH + h) * SS_ + s_) * DH + n * 16 + lo;
      O[off] = acc[n][v];
    }
}

// ---------------------------------------------------------------------------
// Kernel 2: in-place LayerNorm over rows of 64.  One wave per row.
// ---------------------------------------------------------------------------
__global__ __launch_bounds__(256) void ln_rows64_kernel(
    float* __restrict__ T, const float* __restrict__ g,
    const float* __restrict__ b, int nrows) {
  const int wave = threadIdx.x >> 5, lane = threadIdx.x & 31;
  const int row = blockIdx.x * 8 + wave;
  if (row >= nrows) return;
  float* p = T + (size_t)row * DH;
  float2 xv = *(float2*)(p + lane * 2);
  float s = xv.x + xv.y;
  float sq = xv.x * xv.x + xv.y * xv.y;
#pragma unroll
  for (int o = 16; o; o >>= 1) {
    s += __shfl_xor(s, o, 32);
    sq += __shfl_xor(sq, o, 32);
  }
  float mean = s * (1.f / 64.f);
  float var = sq * (1.f / 64.f) - mean * mean;
  float inv = rsqrtf(var + 1e-5f);
  float2 r;
  r.x = (xv.x - mean) * inv * g[lane * 2] + b[lane * 2];
  r.y = (xv.y - mean) * inv * g[lane * 2 + 1] + b[lane * 2 + 1];
  *(float2*)(p + lane * 2) = r;
}

// ---------------------------------------------------------------------------
// Kernel 3: fused attention (no softmax).  maps = Qn Kn^T / 8 ; O = maps @ Vn.
// grid = (4 q-tiles of 128, H, B); 256 threads = 8 waves x 16 q-rows.
// Q/K tiles staged async; V tile staged transposed.
// ---------------------------------------------------------------------------
__global__ __launch_bounds__(256) void attn_wmma_kernel(
    const float* __restrict__ Q, const float* __restrict__ K,
    const float* __restrict__ V, float* __restrict__ maps,
    float* __restrict__ O) {
  const int qt = blockIdx.x;
  const int h = blockIdx.y, b = blockIdx.z;
  const size_t base = (((size_t)b * HH + h) * SS_) * DH;
  const float* Qp = Q + base;
  const float* Kp = K + base;
  const float* Vp = V + base;

  __shared__ __align__(16) float Qs[128][68];
  __shared__ __align__(16) float Ks[16][68];
  __shared__ __align__(16) float VsT[64][20];  // transposed: VsT[n][k]
  __shared__ __align__(16) float Sc[8][16][20];

  const int tid = threadIdx.x;
  const int wave = tid >> 5, lane = tid & 31;
  const int hi = lane >> 4, lo = lane & 15;
  const int qrow = wave * 16;

  // resident Q tile: 128x64 floats = 2048 x 16B, 8 per thread, async DMA
  for (int i = tid; i < 2048; i += 256) {
    int r = i >> 4, c4 = (i & 15) << 2;
    async_copy16(Qp + (size_t)(qt * 128 + r) * DH + c4, &Qs[r][c4]);
  }
  async_wait0();

  v8f oacc[4] = {};
  float* mapsRow = maps + ((size_t)(h * BB + b) * SS_ + qt * 128) * SS_;

  for (int t0 = 0; t0 < SS_; t0 += 16) {
    __syncthreads();
    {
      int r = tid >> 4, c4 = (tid & 15) << 2;
      // K tile 16x64 row-major: async DMA (1 chunk per thread)
      async_copy16(Kp + (size_t)(t0 + r) * DH + c4, &Ks[r][c4]);
      // V tile transposed into VsT[n][k]
      float4 vv = *(const float4*)(Vp + (size_t)(t0 + r) * DH + c4);
      VsT[c4 + 0][r] = vv.x;
      VsT[c4 + 1][r] = vv.y;
      VsT[c4 + 2][r] = vv.z;
      VsT[c4 + 3][r] = vv.w;
    }
    async_wait0();
    __syncthreads();

    // scores tile [16x16]: A = Q rows, B = Kn^T (both K-contiguous in LDS)
    v8f sc = {};
#pragma unroll
    for (int kk = 0; kk < DH; kk += 4) {
      v2f a = ld2(&Qs[qrow + lo][kk + 2 * hi]);
      v2f bf = ld2(&Ks[lo][kk + 2 * hi]);  // B[k][n] = Kn[t0+n][k]
      sc = wmma4(a, bf, sc);
    }

    // scale, emit maps, stash tile for second GEMM
#pragma unroll
    for (int v = 0; v < 8; ++v) {
      float val = sc[v] * 0.125f;  // 1/sqrt(dh)
      int M = v + 8 * hi;
      mapsRow[(size_t)(qrow + M) * SS_ + t0 + lo] = val;
      Sc[wave][M][lo] = val;
    }
    __syncthreads();

    // O += scores @ V  (K dim = 16)
#pragma unroll
    for (int kk = 0; kk < 16; kk += 4) {
      v2f a = ld2(&Sc[wave][lo][kk + 2 * hi]);
#pragma unroll
      for (int n = 0; n < 4; ++n) {
        v2f bf = ld2(&VsT[n * 16 + lo][kk + 2 * hi]);
        oacc[n] = wmma4(a, bf, oacc[n]);
      }
    }
  }

  // O layout: [B, S, H*dh]
#pragma unroll
  for (int n = 0; n < 4; ++n)
#pragma unroll
    for (int v = 0; v < 8; ++v) {
      int srow = qt * 128 + qrow + v + 8 * hi;
      O[((size_t)b * SS_ + srow) * DD + h * DH + n * 16 + lo] = oacc[n][v];
    }
}

// ---------------------------------------------------------------------------
// Kernel 4: out = LayerNorm(X + Y) over D=1024.  One block per row.
// ---------------------------------------------------------------------------
__global__ __launch_bounds__(256) void ln_residual_kernel(
    const float* __restrict__ X, const float* __restrict__ Y,
    const float* __restrict__ g, const float* __restrict__ bta,
    float* __restrict__ out) {
  __shared__ float red[16];
  const int row = blockIdx.x, tid = threadIdx.x;
  const float* xp = X + (size_t)row * DD;
  const float* yp = Y + (size_t)row * DD;
  float4 xv = *(const float4*)(xp + tid * 4);
  float4 yv = *(const float4*)(yp + tid * 4);
  float vals[4] = {xv.x + yv.x, xv.y + yv.y, xv.z + yv.z, xv.w + yv.w};
  float s = 0.f, sq = 0.f;
#pragma unroll
  for (int i = 0; i < 4; ++i) {
    s += vals[i];
    sq += vals[i] * vals[i];
  }
#pragma unroll
  for (int o = 16; o; o >>= 1) {
    s += __shfl_xor(s, o, 32);
    sq += __shfl_xor(sq, o, 32);
  }
  const int wave = tid >> 5, lane = tid & 31;
  if (lane == 0) {
    red[wave] = s;
    red[8 + wave] = sq;
  }
  __syncthreads();
  float ts = 0.f, tsq = 0.f;
#pragma unroll
  for (int i = 0; i < 8; ++i) {
    ts += red[i];
    tsq += red[8 + i];
  }
  float mean = ts * (1.f / 1024.f);
  float var = tsq * (1.f / 1024.f) - mean * mean;
  float inv = rsqrtf(var + 1e-5f);
  float4 r;
  r.x = (vals[0] - mean) * inv * g[tid * 4 + 0] + bta[tid * 4 + 0];
  r.y = (vals[1] - mean) * inv * g[tid * 4 + 1] + bta[tid * 4 + 1];
  r.z = (vals[2] - mean) * inv * g[tid * 4 + 2] + bta[tid * 4 + 2];
  r.w = (vals[3] - mean) * inv * g[tid * 4 + 3] + bta[tid * 4 + 3];
  *(float4*)(out + (size_t)row * DD + tid * 4) = r;
}

// ---------------------------------------------------------------------------
// Kernel 5: MLP layer: out = relu(A[8192,1024] @ W[1024,1024] + bias).
// grid = (64 Mtiles, 16 Ntiles-of-64); 256 threads = 8 waves.
// ---------------------------------------------------------------------------
__global__ __launch_bounds__(256) void mlp_wmma_kernel(
    const float* __restrict__ A, const float* __restrict__ W,
    const float* __restrict__ bias, float* __restrict__ out) {
  const int mt = blockIdx.x;
  const int n0 = blockIdx.y * 64;

  __shared__ __align__(16) float As[128][20];
  __shared__ __align__(16) float BsT[64][20];  // transposed: BsT[n][k]

  const int tid = threadIdx.x;
  const int wave = tid >> 5, lane = tid & 31;
  const int hi = lane >> 4, lo = lane & 15;
  const int m0 = wave * 16;

  v8f acc[4] = {};
  const float* Arow = A + (size_t)mt * 128 * DD;

  for (int k0 = 0; k0 < DD; k0 += 16) {
    __syncthreads();
    for (int i = tid; i < 512; i += 256) {
      int r = i >> 2, c4 = (i & 3) << 2;
      async_copy16(Arow + (size_t)r * DD + k0 + c4, &As[r][c4]);
    }
    {
      int r = tid >> 4, c4 = (tid & 15) << 2;
      float4 w4 = *(const float4*)(W + (size_t)(k0 + r) * DD + n0 + c4);
      BsT[c4 + 0][r] = w4.x;
      BsT[c4 + 1][r] = w4.y;
      BsT[c4 + 2][r] = w4.z;
      BsT[c4 + 3][r] = w4.w;
    }
    async_wait0();
    __syncthreads();

    for (int kk = 0; kk < 16; kk += 4) {
      v2f a = ld2(&As[m0 + lo][kk + 2 * hi]);
#pragma unroll
      for (int n = 0; n < 4; ++n) {
        v2f b = ld2(&BsT[n * 16 + lo][kk + 2 * hi]);
        acc[n] = wmma4(a, b, acc[n]);
      }
    }
  }

#pragma unroll
  for (int n = 0; n < 4; ++n)
#pragma unroll
    for (int v = 0; v < 8; ++v) {
      int gm = mt * 128 + m0 + v + 8 * hi;
      float c = acc[n][v] + bias[n0 + n * 16 + lo];
      out[(size_t)gm * DD + n0 + n * 16 + lo] = fmaxf(c, 0.f);
    }
}

// ---------------------------------------------------------------------------
// Host driver
// ---------------------------------------------------------------------------
extern "C" void kernel_launch(void* const* d_in, const int* in_sizes, int n_in,
                              void* d_out, int out_size, void* d_ws,
                              size_t ws_size, hipStream_t stream) {
  (void)in_sizes;
  (void)n_in;
  (void)out_size;
  (void)ws_size;
  const int NB = 2, L = 2;
  const size_t M = (size_t)BB * SS_;
  const size_t nElem = M * DD;
  const size_t mapsPerBlock = (size_t)HH * BB * SS_ * SS_;

  const float* x_in = (const float*)d_in[0];
  const float* Wq = (const float*)d_in[1];
  const float* Wk = (const float*)d_in[2];
  const float* Wv = (const float*)d_in[3];
  const float* qg = (const float*)d_in[4];
  const float* qb = (const float*)d_in[5];
  const float* kg = (const float*)d_in[6];
  const float* kb = (const float*)d_in[7];
  const float* vg = (const float*)d_in[8];
  const float* vb = (const float*)d_in[9];
  const float* l0g = (const float*)d_in[10];
  const float* l0b = (const float*)d_in[11];
  const float* l1g = (const float*)d_in[12];
  const float* l1b = (const float*)d_in[13];
  const float* mW = (const float*)d_in[14];
  const float* mB = (const float*)d_in[15];

  float* out_x = (float*)d_out;
  float* out_maps = out_x + nElem;

  float* ws = (float*)d_ws;
  float* qn = ws;
  float* kn = qn + nElem;
  float* vn = kn + nElem;
  float* ob = vn + nElem;
  float* nw = ob + nElem;
  float* h0 = nw + nElem;
  float* h1 = h0 + nElem;
  float* xb = h1 + nElem;

  const int headRows = BB * HH * SS_;

  const float* xcur = x_in;
  for (int i = 0; i < NB; ++i) {
    const size_t wOff = (size_t)i * HH * DD * DH;
    proj_wmma_kernel<<<dim3(64, HH, 3), 256, 0, stream>>>(
        xcur, Wq + wOff, Wk + wOff, Wv + wOff, qn, kn, vn);

    ln_rows64_kernel<<<headRows / 8, 256, 0, stream>>>(qn, qg + i * DH,
                                                       qb + i * DH, headRows);
    ln_rows64_kernel<<<headRows / 8, 256, 0, stream>>>(kn, kg + i * DH,
                                                       kb + i * DH, headRows);
    ln_rows64_kernel<<<headRows / 8, 256, 0, stream>>>(vn, vg + i * DH,
                                                       vb + i * DH, headRows);

    attn_wmma_kernel<<<dim3(4, HH, BB), 256, 0, stream>>>(
        qn, kn, vn, out_maps + (size_t)i * mapsPerBlock, ob);

    ln_residual_kernel<<<M, 256, 0, stream>>>(xcur, ob, l0g + (size_t)i * DD,
                                              l0b + (size_t)i * DD, nw);

    mlp_wmma_kernel<<<dim3(64, 16), 256, 0, stream>>>(
        nw, mW + ((size_t)i * L + 0) * DD * DD, mB + ((size_t)i * L + 0) * DD,
        h0);
    mlp_wmma_kernel<<<dim3(64, 16), 256, 0, stream>>>(
        h0, mW + ((size_t)i * L + 1) * DD * DD, mB + ((size_t)i * L + 1) * DD,
        h1);

    float* xnext = (i == NB - 1) ? out_x : xb;
    ln_residual_kernel<<<M, 256, 0, stream>>>(nw, h1, l1g + (size_t)i * DD,
                                              l1b + (size_t)i * DD, xnext);
    xcur = xnext;
  }
}